// ContinuousNormalizingFlowMultiTemperature_45707041964590
// MI455X (gfx1250) — compile-verified
//
#include <hip/hip_runtime.h>
#include <math.h>

// ---------------------------------------------------------------------------
// Continuous Normalizing Flow for MI455X (gfx1250, wave32, WMMA).
//
// One workgroup = 16 batch rows; integrates the full ODE (64 RK4 steps)
// entirely in LDS/registers. Matrix math: V_WMMA_F32_16X16X4_F32 (full fp32,
// required by rtol=1e-5). Divergence via analytic trace:
//   div_i = sum_j (1 - h_ij^2) * c_j,  c_j = sum_{k<D} W1[k,j] * W2[j,k].
//
// v4 changes vs v3:
//  * K-pair-packed LDS layouts: every WMMA fragment (A and B) is now a single
//    aligned ds_load_b64 straight into the operand register pair -- kills the
//    v_mov/v_dual_mov fragment-assembly shuffles seen in v3 asm.
//  * Strides: XS=36/HS=260 (even; in float2 units == 2 mod 4 -> lower lane
//    half on even double-banks, upper on odd -> conflict-free b64 loads;
//    8*260 == 32 mod 64 -> conflict-free tanh writeback). W1/W2 pair-row
//    strides == 16 mod 32 (float2) -> disjoint bank windows per lane half.
// ---------------------------------------------------------------------------

typedef __attribute__((ext_vector_type(2))) float v2f;
typedef __attribute__((ext_vector_type(8))) float v8f;

#define B_TOT    512
#define D        32
#define H        256
#define BR       16         // batch rows per workgroup
#define NTHREADS 256        // 8 waves (wave32)
#define NSTEP    64         // RK4 steps; global error ~1e-7 << rtol 1e-5

// LDS strides (floats).
#define XS   36             // x-like rows [BR][XS]
#define HS   260            // h rows [BR][HS]
#define W1F  544            // W1 pair-rows: 16 k2-rows x 544 (= 2*272, 272%32==16)
#define W2F  96             // W2 pair-rows: 128 k2-rows x 96  (= 2*48,  48%32==16)
#define G2S  36             // GEMM2 partial rows: 8*36 == 32 mod 64

// Branch-free tanh: tanh(z) = sign(z) * (1 - 2/(exp(2|z|) + 1)).
// v_exp_f32 is exp2, so scale by 2*log2(e). Overflow saturates correctly.
__device__ __forceinline__ float fast_tanh(float z) {
    const float TWO_LOG2E = 2.88539008177792681472f;   // 2*log2(e)
    const float a = __builtin_fabsf(z);
    const float e = __builtin_amdgcn_exp2f(a * TWO_LOG2E);   // exp(2|z|)
    const float r = __builtin_amdgcn_rcpf(e + 1.0f);
    const float t = __builtin_fmaf(-2.0f, r, 1.0f);          // tanh(|z|)
    return __builtin_copysignf(t, z);
}

__device__ __forceinline__ v2f lds_v2f(const float* p) {
    return *(const v2f*)p;   // 8B-aligned by construction -> ds_load_b64
}

// Core MLP evaluation. Pre: b1eff & xin valid, synced. Post: outputs valid, synced.
__device__ __forceinline__ void eval_core(
    const float* __restrict__ xin,    // [BR*XS]   LDS (pairs along D)
    float* __restrict__ hbuf,         // [BR*HS]   LDS
    float* __restrict__ g2part,       // [4*BR*G2S] LDS
    float* __restrict__ part,         // [BR*16]
    float* __restrict__ epart,        // [BR*16]
    const float* __restrict__ w1p,    // [16*W1F]  K-pair-packed W1[0:32,:]
    const float* __restrict__ w2p,    // [128*W2F] K-pair-packed W2
    const float* __restrict__ cdiag,  // [H]
    const float* __restrict__ b1eff,  // [H]
    int tid)
{
    const int wid  = tid >> 5;
    const int lane = tid & 31;
    const int half = lane >> 4;   // 16x4 f32 fragments: upper lanes hold K+2 / M+8
    const int lr   = lane & 15;

    // ---- GEMM1: z = x @ W1[0:32,:]  (16x256, K=32 -> 8 WMMA / tile,
    //      8 waves x 2 n-tiles). Writeback applies bias + tanh.
#pragma unroll
    for (int it = 0; it < 2; ++it) {
        const int n0 = (wid * 2 + it) * 16;
        v8f acc = {};
#pragma unroll
        for (int k0 = 0; k0 < 8; ++k0) {
            const int kb = k0 * 4 + 2 * half;          // even
            const v2f a = lds_v2f(&xin[lr * XS + kb]);                 // A[lr][kb..kb+1]
            const v2f b = lds_v2f(&w1p[(2 * k0 + half) * W1F + 2 * (n0 + lr)]); // B[kb..kb+1][n]
            acc = __builtin_amdgcn_wmma_f32_16x16x4_f32(
                      false, a, false, b, (short)0, acc, false, false);
        }
#pragma unroll
        for (int v = 0; v < 8; ++v) {
            const int row = v + 8 * half;              // C/D: M = v (+8 upper half)
            const int col = n0 + lr;
            hbuf[row * HS + col] = fast_tanh(acc[v] + b1eff[col]);
        }
    }
    __syncthreads();

    // ---- GEMM2 split-K: f_partial = h @ W2.  Wave w: n-tile (w&1),
    //      K-segment (w>>1) of 64 -> 16 dependent WMMAs per wave.
    {
        const int n0 = (wid & 1) * 16;
        const int s  = wid >> 1;
        v8f acc = {};
#pragma unroll
        for (int k0i = 0; k0i < 16; ++k0i) {
            const int kk = s * 16 + k0i;
            const int kb = kk * 4 + 2 * half;          // even
            const v2f a = lds_v2f(&hbuf[lr * HS + kb]);
            const v2f b = lds_v2f(&w2p[(2 * kk + half) * W2F + 2 * (n0 + lr)]);
            acc = __builtin_amdgcn_wmma_f32_16x16x4_f32(
                      false, a, false, b, (short)0, acc, false, false);
        }
#pragma unroll
        for (int v = 0; v < 8; ++v) {
            const int row = v + 8 * half;
            g2part[(s * BR + row) * G2S + n0 + lr] = acc[v];
        }
    }

    // ---- divergence & energy partials (VALU; co-executes with WMMA tail)
    {
        const int r = tid >> 4;
        const int sub = tid & 15;
        float dsum = 0.f;
#pragma unroll
        for (int j = sub; j < H; j += 16) {
            const float hv = hbuf[r * HS + j];
            dsum += (1.f - hv * hv) * cdiag[j];
        }
        part[tid] = dsum;
        const float xa = xin[r * XS + sub];
        const float xb = xin[r * XS + sub + 16];
        epart[tid] = 0.5f * (xa * xa + xb * xb);
    }
    __syncthreads();
}

__global__ __launch_bounds__(NTHREADS)
void cnf_rk4_kernel(const float* __restrict__ x,
                    const float* __restrict__ W1,
                    const float* __restrict__ b1,
                    const float* __restrict__ W2,
                    const float* __restrict__ b2,
                    float* __restrict__ out)
{
    __shared__ __align__(16) float w1p[16 * W1F];     // W1[0:32,:] K-pair packed
    __shared__ __align__(16) float w2p[128 * W2F];    // W2 K-pair packed
    __shared__ float w1time[H];                       // W1[32,:] (time row)
    __shared__ float b1s[H], b2s[D], cdiag[H], b1eff[H];
    __shared__ __align__(16) float hbuf[BR * HS];
    __shared__ __align__(16) float x0[BR * XS], xev[BR * XS], xacc[BR * XS];
    __shared__ float g2part[4 * BR * G2S];
    __shared__ float part[BR * 16], epart[BR * 16];
    __shared__ float acclp[BR], accls[BR], lp0[BR], ls0[BR];

    const int tid = threadIdx.x;
    const int wg  = blockIdx.x;

    // ---- cooperative global -> LDS, packing K-adjacent pairs.
    // W1 global [33][256] row-major: k<32 -> pairs; k==32 -> time row.
    for (int i = tid; i < 32 * H; i += NTHREADS) {
        const int k = i / H, n = i % H;
        w1p[(k >> 1) * W1F + 2 * n + (k & 1)] = W1[i];
    }
    w1time[tid] = W1[32 * H + tid];
    // W2 global [256][32]
    for (int i = tid; i < H * D; i += NTHREADS) {
        const int k = i / D, n = i % D;
        w2p[(k >> 1) * W2F + 2 * n + (k & 1)] = W2[i];
    }
    b1s[tid] = b1[tid];
    if (tid < D) b2s[tid] = b2[tid];
    for (int i = tid; i < BR * D; i += NTHREADS)
        x0[(i / D) * XS + (i % D)] = x[wg * BR * D + i];
    if (tid < BR) { lp0[tid] = 0.f; ls0[tid] = 0.f; }
    __syncthreads();

    // Per-thread register cache for the bias update (one FMA + ds_store/stage)
    const float b1r = b1s[tid];
    const float w1t = w1time[tid];

    // cdiag[j] = sum_{k<D} W1[k][j] * W2[j][k]; b1eff for t=0 (stage 1)
    {
        float c = 0.f;
        const int j = tid;
#pragma unroll
        for (int k = 0; k < D; ++k)
            c += w1p[(k >> 1) * W1F + 2 * j + (k & 1)]
               * w2p[(j >> 1) * W2F + 2 * k + (j & 1)];
        cdiag[j] = c;
        b1eff[j] = b1r;                      // t = 0
    }
    __syncthreads();

    const float dt = 1.0f / (float)NSTEP;

    for (int step = 0; step < NSTEP; ++step) {
        const float t0 = step * dt;

        // ---------------- RK4 stage 1: k1 = F(x0, t0) ----------------
        eval_core(x0, hbuf, g2part, part, epart, w1p, w2p, cdiag, b1eff, tid);
        for (int i = tid; i < BR * D; i += NTHREADS) {
            const int r = i >> 5, d = i & 31;
            const float fv = b2s[d]
                + g2part[(0 * BR + r) * G2S + d] + g2part[(1 * BR + r) * G2S + d]
                + g2part[(2 * BR + r) * G2S + d] + g2part[(3 * BR + r) * G2S + d];
            xacc[r * XS + d] = fv;
            xev[r * XS + d]  = x0[r * XS + d] + 0.5f * dt * fv;
        }
        if (tid < BR) {
            float ds = 0.f, es = 0.f;
#pragma unroll
            for (int s = 0; s < 16; ++s) { ds += part[tid * 16 + s]; es += epart[tid * 16 + s]; }
            acclp[tid] = -ds;  accls[tid] = es;
        }
        b1eff[tid] = b1r + (t0 + 0.5f * dt) * w1t;   // for stages 2 & 3
        __syncthreads();

        // ---------------- RK4 stage 2 ----------------
        eval_core(xev, hbuf, g2part, part, epart, w1p, w2p, cdiag, b1eff, tid);
        for (int i = tid; i < BR * D; i += NTHREADS) {
            const int r = i >> 5, d = i & 31;
            const float fv = b2s[d]
                + g2part[(0 * BR + r) * G2S + d] + g2part[(1 * BR + r) * G2S + d]
                + g2part[(2 * BR + r) * G2S + d] + g2part[(3 * BR + r) * G2S + d];
            xacc[r * XS + d] += 2.f * fv;
            xev[r * XS + d]   = x0[r * XS + d] + 0.5f * dt * fv;
        }
        if (tid < BR) {
            float ds = 0.f, es = 0.f;
#pragma unroll
            for (int s = 0; s < 16; ++s) { ds += part[tid * 16 + s]; es += epart[tid * 16 + s]; }
            acclp[tid] += -2.f * ds;  accls[tid] += 2.f * es;
        }
        __syncthreads();                              // b1eff unchanged (same t)

        // ---------------- RK4 stage 3 ----------------
        eval_core(xev, hbuf, g2part, part, epart, w1p, w2p, cdiag, b1eff, tid);
        for (int i = tid; i < BR * D; i += NTHREADS) {
            const int r = i >> 5, d = i & 31;
            const float fv = b2s[d]
                + g2part[(0 * BR + r) * G2S + d] + g2part[(1 * BR + r) * G2S + d]
                + g2part[(2 * BR + r) * G2S + d] + g2part[(3 * BR + r) * G2S + d];
            xacc[r * XS + d] += 2.f * fv;
            xev[r * XS + d]   = x0[r * XS + d] + dt * fv;
        }
        if (tid < BR) {
            float ds = 0.f, es = 0.f;
#pragma unroll
            for (int s = 0; s < 16; ++s) { ds += part[tid * 16 + s]; es += epart[tid * 16 + s]; }
            acclp[tid] += -2.f * ds;  accls[tid] += 2.f * es;
        }
        b1eff[tid] = b1r + (t0 + dt) * w1t;           // for stage 4
        __syncthreads();

        // ---------------- RK4 stage 4 + state update ----------------
        eval_core(xev, hbuf, g2part, part, epart, w1p, w2p, cdiag, b1eff, tid);
        for (int i = tid; i < BR * D; i += NTHREADS) {
            const int r = i >> 5, d = i & 31;
            const float fv = b2s[d]
                + g2part[(0 * BR + r) * G2S + d] + g2part[(1 * BR + r) * G2S + d]
                + g2part[(2 * BR + r) * G2S + d] + g2part[(3 * BR + r) * G2S + d];
            x0[r * XS + d] += (dt / 6.f) * (xacc[r * XS + d] + fv);
        }
        if (tid < BR) {
            float ds = 0.f, es = 0.f;
#pragma unroll
            for (int s = 0; s < 16; ++s) { ds += part[tid * 16 + s]; es += epart[tid * 16 + s]; }
            lp0[tid] += (dt / 6.f) * (acclp[tid] - ds);
            ls0[tid] += (dt / 6.f) * (accls[tid] + es);
        }
        b1eff[tid] = b1r + (t0 + dt) * w1t;           // stage 1 of next step
        __syncthreads();
    }

    // ---- write results: [x_final | logp | loss] concatenated flat
    for (int i = tid; i < BR * D; i += NTHREADS)
        out[wg * BR * D + i] = x0[(i >> 5) * XS + (i & 31)];
    if (tid < BR) {
        out[B_TOT * D + wg * BR + tid]         = lp0[tid];
        out[B_TOT * D + B_TOT + wg * BR + tid] = ls0[tid];
    }
}

extern "C" void kernel_launch(void* const* d_in, const int* in_sizes, int n_in,
                              void* d_out, int out_size, void* d_ws, size_t ws_size,
                              hipStream_t stream) {
    const float* x  = (const float*)d_in[0];
    const float* W1 = (const float*)d_in[1];
    const float* b1 = (const float*)d_in[2];
    const float* W2 = (const float*)d_in[3];
    const float* b2 = (const float*)d_in[4];
    float* out = (float*)d_out;
    cnf_rk4_kernel<<<B_TOT / BR, NTHREADS, 0, stream>>>(x, W1, b1, W2, b2, out);
}